// AIDWLayer_72550587564740
// MI455X (gfx1250) — compile-verified
//
#include <hip/hip_runtime.h>
#include <hip/hip_bf16.h>

typedef __attribute__((ext_vector_type(2))) float v2f;
typedef __attribute__((ext_vector_type(8))) float v8f;

// Problem constants (from reference setup_inputs)
#define BATCH 32
#define SEQ   8192
#define NSTA  64
#define OUTF  128

#define MTILE 128            // rows of C per workgroup
#define WPAD  66             // padded LDS stride for W^T (avoids bank conflicts)

__global__ __launch_bounds__(256)
void aidw_wmma_kernel(const float* __restrict__ features,
                      const float* __restrict__ src_locs,
                      const float* __restrict__ tar_loc,
                      const unsigned char* __restrict__ src_masks,
                      const float* __restrict__ linear,
                      float* __restrict__ out) {
    __shared__ float sW[OUTF * WPAD];   // W^T: [o][k], padded stride 66 -> ~33 KB
    __shared__ float sRaw[NSTA];

    const int b    = blockIdx.y;
    const int row0 = blockIdx.x * MTILE;
    const int tid  = threadIdx.x;

    // ---- 1) IDW raw weights for this batch (BETA = 2 -> dist^-2 == 1/d2) ----
    if (tid < NSTA) {
        const int n = tid;
        const float tx = tar_loc[b * 2 + 0];
        const float ty = tar_loc[b * 2 + 1];
        const float dx = src_locs[(b * NSTA + n) * 2 + 0] - tx;
        const float dy = src_locs[(b * NSTA + n) * 2 + 1] - ty;
        const float d2 = dx * dx + dy * dy;
        sRaw[n] = src_masks[b * NSTA + n] ? (1.0f / d2) : 0.0f;
    }
    __syncthreads();

    // Every thread privately sums the 64 raw weights (LDS broadcast reads).
    float total = 0.0f;
    #pragma unroll
    for (int n = 0; n < NSTA; ++n) total += sRaw[n];
    const float invT = 1.0f / total;

    // ---- 2) Build W^T[o][k] = scores[k] * linear[k][o] in LDS ----
    for (int i = tid; i < NSTA * OUTF; i += 256) {
        const int n = i >> 7;        // station (k)
        const int o = i & (OUTF - 1);
        sW[o * WPAD + n] = sRaw[n] * invT * linear[i];   // linear is row-major [n][o]
    }
    __syncthreads();

    // ---- 3) GEMM: C[128x128] = A[128x64] x W[64x128] via V_WMMA_F32_16X16X4_F32 ----
    const int wave = tid >> 5;          // 0..7 : 16-row strip within the 128-row tile
    const int lane = tid & 31;
    const int nlo  = lane & 15;         // M (for A) / N (for B,C) within tile
    const int koff = (lane >> 4) * 2;   // upper half-wave handles K+2,K+3

    const float* Arow = features + ((size_t)b * SEQ + row0 + wave * 16 + nlo) * NSTA;

    v8f acc[8];
    #pragma unroll
    for (int ct = 0; ct < 8; ++ct) acc[ct] = (v8f){0.f,0.f,0.f,0.f,0.f,0.f,0.f,0.f};

    #pragma unroll
    for (int k = 0; k < NSTA; k += 4) {
        // A fragment 16x4: lane holds A[m][k+koff], A[m][k+koff+1]
        const v2f a = *(const v2f*)(Arow + k + koff);
        #pragma unroll
        for (int ct = 0; ct < 8; ++ct) {
            // B fragment 4x16: lane holds W[k+koff][n], W[k+koff+1][n]
            // stored transposed in LDS -> consecutive, 8B-aligned (index even)
            const v2f bb = *(const v2f*)(&sW[(ct * 16 + nlo) * WPAD + k + koff]);
            acc[ct] = __builtin_amdgcn_wmma_f32_16x16x4_f32(
                /*neg_a=*/false, a, /*neg_b=*/false, bb,
                /*c_mod=*/(short)0, acc[ct],
                /*reuse_a=*/false, /*reuse_b=*/false);
        }
    }

    // ---- 4) Store C: VGPR v <-> row v + 8*(lane/16); lane%16 <-> column ----
    float* Crow = out + ((size_t)b * SEQ + row0 + wave * 16) * OUTF;
    const int mhalf = (lane >> 4) * 8;
    #pragma unroll
    for (int ct = 0; ct < 8; ++ct) {
        #pragma unroll
        for (int v = 0; v < 8; ++v) {
            Crow[(mhalf + v) * OUTF + ct * 16 + nlo] = acc[ct][v];
        }
    }
}

extern "C" void kernel_launch(void* const* d_in, const int* in_sizes, int n_in,
                              void* d_out, int out_size, void* d_ws, size_t ws_size,
                              hipStream_t stream) {
    const float*         features  = (const float*)d_in[0];          // (B,S,N)
    const float*         src_locs  = (const float*)d_in[1];          // (B,N,2)
    const float*         tar_loc   = (const float*)d_in[2];          // (B,2)
    const unsigned char* src_masks = (const unsigned char*)d_in[3];  // (B,N) bool
    const float*         linear    = (const float*)d_in[4];          // (N,O)
    float*               out       = (float*)d_out;                  // (B,S,O)

    dim3 grid(SEQ / MTILE, BATCH);   // (64, 32)
    dim3 block(256);
    aidw_wmma_kernel<<<grid, block, 0, stream>>>(features, src_locs, tar_loc,
                                                 src_masks, linear, out);
}